// Tacotron_84189948936259
// MI455X (gfx1250) — compile-verified
//
#include <hip/hip_runtime.h>

// ---------------------------------------------------------------------------
// Tacotron forward for MI455X (gfx1250, wave32, WMMA).
// All GEMMs use V_WMMA_F32_16X16X4_F32 (native fp32 matrix path, keeps the
// reference's fp32 numerics). Parallel GEMM/conv kernels compute a 16x64 C
// strip per wave (4 WMMA per A fragment), with all four B fragments loaded
// before the WMMA chain so the compiler emits clause'd loads + staggered
// s_wait_loadcnt instead of full drains. Encoder bi-GRU and the attention
// decoder are persistent single-workgroup kernels with state held in LDS
// (decoder uses ~304KB of the 320KB/WGP LDS, requested as dynamic LDS).
// Workspace requirement: ~88 MB.
// ---------------------------------------------------------------------------

typedef __attribute__((ext_vector_type(2))) float v2f;
typedef __attribute__((ext_vector_type(8))) float v8f;

#define ACT_NONE 0
#define ACT_RELU 1
#define ACT_SIG  2
#define ACT_TANH 3

__device__ __forceinline__ v8f wmma4(v2f a, v2f b, v8f c) {
  // D = A(16x4) * B(4x16) + C(16x16), fp32 throughout.
  return __builtin_amdgcn_wmma_f32_16x16x4_f32(false, a, false, b, (short)0, c,
                                               false, false);
}

__device__ __forceinline__ float actf(float v, int act) {
  if (act == ACT_RELU) return fmaxf(v, 0.f);
  if (act == ACT_SIG)  return 1.f / (1.f + __expf(-v));
  if (act == ACT_TANH) return tanhf(v);
  return v;
}

// ---------------------------------------------------------------------------
// Generic GEMM: C[M,N] = act(A[M,K] @ B[K,N] + bias) (+ residual)
// Requires M%16==0, N%64==0, K%4==0 (true at every call site).
// Optional rowidx gathers A rows (fused embedding lookup).
// One wave computes a 16x64 strip (4 accumulators); 4 waves / block.
// ---------------------------------------------------------------------------
__global__ void gemm_k(const float* __restrict__ A, int lda,
                       const int* __restrict__ rowidx,
                       const float* __restrict__ B, int ldb,
                       const float* __restrict__ bias,
                       float* __restrict__ C, int ldc,
                       int M, int N, int K, int act,
                       const float* __restrict__ res, int ldr) {
  int wave = threadIdx.x >> 5, lane = threadIdx.x & 31;
  int half = lane >> 4, l = lane & 15;
  int nblk = N >> 6;
  int unit = blockIdx.x * 4 + wave;
  if (unit >= (M >> 4) * nblk) return;     // wave-uniform exit
  int tm = (unit / nblk) << 4, tn = (unit % nblk) << 6;
  int m = tm + l;
  const float* Ap = A + (long)(rowidx ? rowidx[m] : m) * lda + (half << 1);
  const float* b0 = B + tn + l + (long)(half << 1) * ldb;
  const long bstep = 4L * ldb;
  const int kpf = K - 32;                  // uniform prefetch horizon
  v8f acc[4] = {};
  for (int k = 0; k < K; k += 4) {
    v2f a; a.x = Ap[k]; a.y = Ap[k + 1];   // global_load_b64
    if (k < kpf)                           // scalar branch (k, kpf uniform)
      __builtin_prefetch(b0 + 8 * bstep, 0, 1);  // global_prefetch_b8
    v2f b[4];
#pragma unroll
    for (int j = 0; j < 4; ++j) {          // all loads issued first (clause)
      b[j].x = b0[16 * j]; b[j].y = b0[16 * j + ldb];
    }
#pragma unroll
    for (int j = 0; j < 4; ++j)            // then the 4-deep WMMA chain
      acc[j] = wmma4(a, b[j], acc[j]);
    b0 += bstep;                           // strength-reduced B row pointer
  }
#pragma unroll
  for (int j = 0; j < 4; ++j) {
    int n = tn + 16 * j + l;
    float bv = bias ? bias[n] : 0.f;
#pragma unroll
    for (int r = 0; r < 8; ++r) {
      int row = tm + r + (half << 3);
      float v = actf(acc[j][r] + bv, act);
      if (res) v += res[(long)row * ldr + n];
      C[(long)row * ldc + n] = v;
    }
  }
}

// ---------------------------------------------------------------------------
// Conv1d-as-GEMM over (B=32, T=256, Cin) with 'SAME' padding.
// Cin = 1<<cinShift; Cout%64==0. Padding handled by clamped-address loads +
// select-0 (keeps loads unpredicated). Optional fused max-pool-over-(t,t+1)
// on the input (proj1 reading the conv bank) and fused residual on output.
// One wave computes a 16x64 strip.
// ---------------------------------------------------------------------------
__global__ void conv_gemm_k(const float* __restrict__ X, int cinShift,
                            const float* __restrict__ W, int KW,
                            const float* __restrict__ bias,
                            float* __restrict__ Y, int ldy, int ycol0,
                            int Cout, int act,
                            const float* __restrict__ res, int ldr,
                            int maxpool) {
  const int Cin = 1 << cinShift;
  int wave = threadIdx.x >> 5, lane = threadIdx.x & 31;
  int half = lane >> 4, l = lane & 15;
  int nblk = Cout >> 6;
  int unit = blockIdx.x * 4 + wave;
  if (unit >= (nblk << 9)) return;
  int tm = (unit / nblk) << 4, tn = (unit % nblk) << 6;
  int m = tm + l;
  int b = m >> 8, t = m & 255;
  int pad = (KW - 1) >> 1;
  int K = KW << cinShift;
  const float* b0 = W + tn + l + (long)(half << 1) * Cout;
  const long wstep = 4L * (long)Cout;
  const int kpf = K - 32;
  v8f acc[4] = {};
  for (int k = 0; k < K; k += 4) {
    int ka = k + (half << 1);
    int kw = ka >> cinShift, c = ka & (Cin - 1);
    int tt = t + kw - pad;
    int tcl = tt < 0 ? 0 : (tt > 255 ? 255 : tt);
    long base = ((long)((b << 8) + tcl) << cinShift) + c;
    float ax = X[base], ay = X[base + 1];
    if (maxpool) {
      long b2 = ((long)((b << 8) + (tcl < 255 ? tcl + 1 : 255)) << cinShift) + c;
      ax = fmaxf(ax, X[b2]); ay = fmaxf(ay, X[b2 + 1]);
    }
    bool inb = (tt >= 0) && (tt < 256);
    v2f a; a.x = inb ? ax : 0.f; a.y = inb ? ay : 0.f;
    if (k < kpf)
      __builtin_prefetch(b0 + 8 * wstep, 0, 1);
    v2f bb[4];
#pragma unroll
    for (int j = 0; j < 4; ++j) {
      bb[j].x = b0[16 * j]; bb[j].y = b0[16 * j + Cout];
    }
#pragma unroll
    for (int j = 0; j < 4; ++j)
      acc[j] = wmma4(a, bb[j], acc[j]);
    b0 += wstep;
  }
#pragma unroll
  for (int j = 0; j < 4; ++j) {
    int n = tn + 16 * j + l;
    float bvv = bias ? bias[n] : 0.f;
#pragma unroll
    for (int r = 0; r < 8; ++r) {
      int row = tm + r + (half << 3);
      float v = actf(acc[j][r] + bvv, act);
      if (res) v += res[(long)row * ldr + n];
      Y[(long)row * ldy + ycol0 + n] = v;
    }
  }
}

// h = Hh*T + h*(1-T)
__global__ void hw_combine_k(float* __restrict__ h, const float* __restrict__ Hh,
                             const float* __restrict__ Tt, int n) {
  int i = blockIdx.x * blockDim.x + threadIdx.x;
  if (i < n) { float t = Tt[i]; h[i] = fmaf(Hh[i] - h[i], t, h[i]); }
}

// ---------------------------------------------------------------------------
// Encoder bi-GRU (H=128), persistent: grid.x = direction (0 fwd, 1 bwd).
// One workgroup of 32 waves does the full 256-step scan; h/gates/candidate
// live in LDS (64KB dynamic). Writes memory (B,T,256) halves.
// ---------------------------------------------------------------------------
struct GruW { const float* Wg; const float* bg; const float* Wc; const float* bc; };

__global__ void __launch_bounds__(1024) enc_gru_k(const float* __restrict__ X,
                                                  GruW wf, GruW wb,
                                                  float* __restrict__ mem) {
  extern __shared__ float sm[];
  float* s_h = sm;            // 32*128
  float* s_g = s_h + 4096;    // 32*256
  float* s_c = s_g + 8192;    // 32*128
  int tid = threadIdx.x;
  int dir = blockIdx.x;
  GruW w = dir ? wb : wf;
  int off = dir ? 128 : 0;
  for (int i = tid; i < 4096; i += 1024) s_h[i] = 0.f;
  __syncthreads();
  int wave = tid >> 5, lane = tid & 31, half = lane >> 4, l = lane & 15;
  for (int s = 0; s < 256; ++s) {
    int t = dir ? 255 - s : s;
    { // gates: (32,256) = [x|h](32,256) @ Wg, sigmoid. 32 tiles, 1 per wave.
      int tm = (wave >> 4) << 4, tn = (wave & 15) << 4;
      int m = tm + l, n = tn + l;
      const float* xrow = X + ((long)((m << 8) + t) << 7);
      const float* bw = w.Wg + n + ((half << 1) << 8);   // row ka, stride 256
      v8f acc = {};
      for (int k = 0; k < 256; k += 4) {
        int ka = k + (half << 1);
        v2f a;
        if (ka < 128) { a.x = xrow[ka]; a.y = xrow[ka + 1]; }
        else { a.x = s_h[(m << 7) + ka - 128]; a.y = s_h[(m << 7) + ka - 127]; }
        v2f b; b.x = bw[0]; b.y = bw[256];
        acc = wmma4(a, b, acc);
        bw += 1024;
      }
      for (int r = 0; r < 8; ++r) {
        int row = tm + r + (half << 3);
        float v = acc[r] + w.bg[n];
        s_g[(row << 8) + n] = 1.f / (1.f + __expf(-v));
      }
    }
    __syncthreads();
    if (wave < 16) { // candidate: (32,128) = [x|r*h] @ Wc, tanh. 16 tiles.
      int tm = (wave >> 3) << 4, tn = (wave & 7) << 4;
      int m = tm + l, n = tn + l;
      const float* xrow = X + ((long)((m << 8) + t) << 7);
      const float* bw = w.Wc + n + ((half << 1) << 7);   // row ka, stride 128
      v8f acc = {};
      for (int k = 0; k < 256; k += 4) {
        int ka = k + (half << 1);
        v2f a;
        if (ka < 128) { a.x = xrow[ka]; a.y = xrow[ka + 1]; }
        else {
          int j = ka - 128;
          a.x = s_g[(m << 8) + j] * s_h[(m << 7) + j];
          a.y = s_g[(m << 8) + j + 1] * s_h[(m << 7) + j + 1];
        }
        v2f b; b.x = bw[0]; b.y = bw[128];
        acc = wmma4(a, b, acc);
        bw += 512;
      }
      for (int r = 0; r < 8; ++r) {
        int row = tm + r + (half << 3);
        s_c[(row << 7) + n] = tanhf(acc[r] + w.bc[n]);
      }
    }
    __syncthreads();
    for (int i = tid; i < 4096; i += 1024) { // h = z*h + (1-z)*c; emit
      int m = i >> 7, j = i & 127;
      float z = s_g[(m << 8) + 128 + j];
      float hn = z * s_h[i] + (1.f - z) * s_c[i];
      s_h[i] = hn;
      mem[((long)((m << 8) + t) << 8) + off + j] = hn;
    }
    __syncthreads();
  }
}

// ---------------------------------------------------------------------------
// Decoder helpers: in-LDS GEMM with up to 3 concatenated A segments; the
// third segment can be multiplied elementwise by r (gates[:, :256]).
// Segment widths are multiples of 4, so a lane's (ka,ka+1) never straddles.
// ---------------------------------------------------------------------------
__device__ __forceinline__ v2f dec_apair(const float* A0, int ld0, int n0,
                                         const float* A1, int ld1, int n1,
                                         const float* A2, int ld2,
                                         const float* rmul, int m, int ka) {
  v2f a;
  if (ka < n0) {
    const float* p = A0 + m * ld0 + ka; a.x = p[0]; a.y = p[1];
  } else if (ka - n0 < n1) {
    const float* p = A1 + m * ld1 + (ka - n0); a.x = p[0]; a.y = p[1];
  } else {
    int j = ka - n0 - n1;
    const float* p = A2 + m * ld2 + j; a.x = p[0]; a.y = p[1];
    if (rmul) { const float* r = rmul + (m << 9) + j; a.x *= r[0]; a.y *= r[1]; }
  }
  return a;
}

__device__ __forceinline__ void dk_gemm(const float* A0, int ld0, int n0,
                                        const float* A1, int ld1, int n1,
                                        const float* A2, int ld2,
                                        const float* rmul,
                                        const float* B, int ldb,
                                        const float* bias,
                                        float* C, int ldc,
                                        int N, int K, int act, int tid) {
  int wave = tid >> 5, lane = tid & 31, half = lane >> 4, l = lane & 15;
  int ntn = N >> 4;
  int ntiles = ntn << 1;  // M = 32
  for (int tile = wave; tile < ntiles; tile += 32) {
    int tm = (tile / ntn) << 4, tn = (tile % ntn) << 4;
    int m = tm + l, n = tn + l;
    const float* Bp = B + n + (half << 1) * ldb;
    v8f acc = {};
    for (int k = 0; k < K; k += 4) {
      int ka = k + (half << 1);
      v2f a = dec_apair(A0, ld0, n0, A1, ld1, n1, A2, ld2, rmul, m, ka);
      v2f b; b.x = Bp[0]; b.y = Bp[ldb];
      acc = wmma4(a, b, acc);
      Bp += 4 * ldb;
    }
    for (int r = 0; r < 8; ++r) {
      int row = tm + r + (half << 3);
      float vv = acc[r] + (bias ? bias[n] : 0.f);
      C[row * ldc + n] = actf(vv, act);
    }
  }
}

// GRU (H=256): x = [x0|x1], gates in sg(32x512), candidate in sc(32x256).
__device__ void dec_gru(const float* x0, int ld0, int n0,
                        const float* x1, int ld1, int n1,
                        float* h,
                        const float* Wg, const float* bg,
                        const float* Wc, const float* bc,
                        float* sg, float* sc, float* hnew, int tid) {
  int nin = n0 + n1;
  int K = nin + 256;
  dk_gemm(x0, ld0, n0, x1, ld1, n1, h, 256, nullptr, Wg, 512, bg,
          sg, 512, 512, K, ACT_SIG, tid);
  __syncthreads();
  dk_gemm(x0, ld0, n0, x1, ld1, n1, h, 256, sg, Wc, 256, bc,
          sc, 256, 256, K, ACT_TANH, tid);
  __syncthreads();
  for (int i = tid; i < 8192; i += 1024) {
    int m = i >> 8, j = i & 255;
    float z = sg[(m << 9) + 256 + j];
    hnew[i] = z * h[i] + (1.f - z) * sc[i];
  }
  __syncthreads();
}

struct DecP {
  const float *keys, *mem, *Wq, *v;
  const float *arWg, *arbg, *arWc, *arbc;
  const float *d0Wg, *d0bg, *d0Wc, *d0bc;
  const float *d1Wg, *d1bg, *d1Wc, *d1bc;
  const float *Wp0, *bp0, *Wp1, *bp1;
  const float *Wo, *bo, *Wv, *bv;
  const float *h0;
  float *outs, *alphas;
};

// ---------------------------------------------------------------------------
// Persistent attention decoder: one workgroup, 32 waves, 100 steps.
// All state + temporaries in LDS (~304KB of CDNA5's 320KB/WGP).
// ---------------------------------------------------------------------------
__global__ void __launch_bounds__(1024) decoder_k(DecP p) {
  extern __shared__ float sm[];
  float* s_hatt  = sm;                 // 32x256 (q / h_att)
  float* s_hatt2 = s_hatt  + 8192;     // 32x256 (h_att2)
  float* s_hd0   = s_hatt2 + 8192;     // 32x256
  float* s_hd1   = s_hd0   + 8192;     // 32x256
  float* s_ctx   = s_hd1   + 8192;     // 32x256
  float* s_tmp   = s_ctx   + 8192;     // qW / res0 / res1
  float* s_t2    = s_tmp   + 8192;     // p1a / score / alpha / candidate
  float* s_p1    = s_t2    + 8192;     // 32x128
  float* s_g     = s_p1    + 4096;     // 32x512 gates
  int tid = threadIdx.x, wave = tid >> 5, lane = tid & 31;

  for (int i = tid; i < 8192; i += 1024) {
    s_hatt[i] = p.h0[i]; s_hd0[i] = 0.f; s_hd1[i] = 0.f;
  }
  __syncthreads();

  for (int step = 0; step < 100; ++step) {
    // p1a = relu(h_d1 @ Wp0 + bp0)
    dk_gemm(nullptr, 0, 0, nullptr, 0, 0, s_hd1, 256, nullptr,
            p.Wp0, 256, p.bp0, s_t2, 256, 256, 256, ACT_RELU, tid);
    __syncthreads();
    // p1 = relu(p1a @ Wp1 + bp1)
    dk_gemm(nullptr, 0, 0, nullptr, 0, 0, s_t2, 256, nullptr,
            p.Wp1, 128, p.bp1, s_p1, 128, 128, 256, ACT_RELU, tid);
    __syncthreads();
    // qW = h_att @ Wq
    dk_gemm(nullptr, 0, 0, nullptr, 0, 0, s_hatt, 256, nullptr,
            p.Wq, 256, nullptr, s_tmp, 256, 256, 256, ACT_NONE, tid);
    __syncthreads();
    // score[b,t] = sum_e tanh(keys[b,t,e] + qW[b,e]) * v[e]
    for (int row = wave; row < 8192; row += 32) {
      int b = row >> 8, t = row & 255;
      const float* kr = p.keys + ((long)row << 8);
      float s = 0.f;
      for (int e = lane; e < 256; e += 32)
        s += tanhf(kr[e] + s_tmp[(b << 8) + e]) * p.v[e];
      for (int o = 16; o > 0; o >>= 1) s += __shfl_xor(s, o, 32);
      if (lane == 0) s_t2[(b << 8) + t] = s;
    }
    __syncthreads();
    { // softmax over t (row b = wave), emit alpha
      int b = wave;
      float mx = -1e30f;
      for (int t = lane; t < 256; t += 32) mx = fmaxf(mx, s_t2[(b << 8) + t]);
      for (int o = 16; o > 0; o >>= 1) mx = fmaxf(mx, __shfl_xor(mx, o, 32));
      float sum = 0.f;
      for (int t = lane; t < 256; t += 32) {
        float e = __expf(s_t2[(b << 8) + t] - mx);
        s_t2[(b << 8) + t] = e; sum += e;
      }
      for (int o = 16; o > 0; o >>= 1) sum += __shfl_xor(sum, o, 32);
      float inv = 1.f / sum;
      for (int t = lane; t < 256; t += 32) {
        float a = s_t2[(b << 8) + t] * inv;
        s_t2[(b << 8) + t] = a;
        p.alphas[((long)step << 13) + (b << 8) + t] = a;
      }
    }
    __syncthreads();
    // ctx[b,d] = sum_t alpha[b,t] * mem[b,t,d]
    for (int i = tid; i < 8192; i += 1024) {
      int b = i >> 8, d = i & 255;
      const float* mb = p.mem + ((long)(b << 8) << 8) + d;
      const float* ab = s_t2 + (b << 8);
      float acc = 0.f;
      for (int t = 0; t < 256; ++t) acc += ab[t] * mb[t << 8];
      s_ctx[i] = acc;
    }
    __syncthreads();
    // att_rnn: x=[p1|ctx], h=q -> h_att2
    dec_gru(s_p1, 128, 128, s_ctx, 256, 256, s_hatt,
            p.arWg, p.arbg, p.arWc, p.arbc, s_g, s_t2, s_hatt2, tid);
    // dec0: x=[h_att2|ctx], h=h_d0 (in place)
    dec_gru(s_hatt2, 256, 256, s_ctx, 256, 256, s_hd0,
            p.d0Wg, p.d0bg, p.d0Wc, p.d0bc, s_g, s_t2, s_hd0, tid);
    // res0 = q + h_d0'
    for (int i = tid; i < 8192; i += 1024) s_tmp[i] = s_hatt[i] + s_hd0[i];
    __syncthreads();
    // dec1: x=res0, h=h_d1 (in place)
    dec_gru(s_tmp, 256, 256, nullptr, 0, 0, s_hd1,
            p.d1Wg, p.d1bg, p.d1Wc, p.d1bc, s_g, s_t2, s_hd1, tid);
    // res1 = res0 + h_d1'
    for (int i = tid; i < 8192; i += 1024) s_tmp[i] += s_hd1[i];
    __syncthreads();
    // dense = [res1@Wo+bo, sigmoid(res1@Wv+bv)]  (32x63, VALU dot products)
    for (int i = tid; i < 32 * 63; i += 1024) {
      int b = i / 63, d = i % 63;
      const float* r = s_tmp + (b << 8);
      float acc;
      if (d < 62) {
        acc = p.bo[d];
        for (int e = 0; e < 256; ++e) acc += r[e] * p.Wo[e * 62 + d];
      } else {
        acc = p.bv[0];
        for (int e = 0; e < 256; ++e) acc += r[e] * p.Wv[e];
        acc = 1.f / (1.f + __expf(-acc));
      }
      p.outs[(long)step * 2016 + i] = acc;
    }
    // commit h_att = h_att2
    for (int i = tid; i < 8192; i += 1024) s_hatt[i] = s_hatt2[i];
    __syncthreads();
  }
}

// final[i,b,d] = outs[i/5, b, d]
__global__ void final_repeat_k(float* __restrict__ out,
                               const float* __restrict__ outs) {
  int idx = blockIdx.x * blockDim.x + threadIdx.x;
  if (idx < 500 * 32 * 63) {
    int i = idx / 2016, rem = idx % 2016;
    out[idx] = outs[(i / 5) * 2016 + rem];
  }
}

// ---------------------------------------------------------------------------
// Host orchestration. Param order assumes jax tree_flatten of setup_inputs():
// inputs, outputs, then params dict leaves with keys sorted alphabetically.
// ---------------------------------------------------------------------------
extern "C" void kernel_launch(void* const* d_in, const int* in_sizes, int n_in,
                              void* d_out, int out_size, void* d_ws, size_t ws_size,
                              hipStream_t stream) {
  (void)in_sizes; (void)n_in; (void)out_size; (void)ws_size;
  const int* tok = (const int*)d_in[0];   // (32,256) token ids
  auto F = [&](int i) { return (const float*)d_in[i]; };

  const float *attWm = F(2), *attWq = F(3), *attv = F(4);
  const float *aiW = F(5), *aib = F(6);
  const float *arWc = F(7), *arWg = F(8), *arbc = F(9), *arbg = F(10);
  const float* bankW[16]; for (int i = 0; i < 16; ++i) bankW[i] = F(11 + i);
  const float *d0Wc = F(27), *d0Wg = F(28), *d0bc = F(29), *d0bg = F(30);
  const float *d1Wc = F(31), *d1Wg = F(32), *d1bc = F(33), *d1bg = F(34);
  const float *dpW0 = F(35), *dpb0 = F(36), *dpW1 = F(37), *dpb1 = F(38);
  const float *embed = F(39);
  const float *epW0 = F(40), *epb0 = F(41), *epW1 = F(42), *epb1 = F(43);
  const float *gbWc = F(44), *gbWg = F(45), *gbbc = F(46), *gbbg = F(47);
  const float *gfWc = F(48), *gfWg = F(49), *gfbc = F(50), *gfbg = F(51);
  const float* hwp[16]; for (int i = 0; i < 16; ++i) hwp[i] = F(52 + i);
  const float *WoW = F(68), *Wob = F(69), *WvW = F(70), *Wvb = F(71);
  const float *pj1W = F(72), *pj1b = F(73), *pj2W = F(74), *pj2b = F(75);

  float* ws = (float*)d_ws;                 // peak ~88 MB with overlays
  float* t_h1   = ws;                       // 8192x256 prenet hidden
  float* t_x    = ws + 2097152;             // 8192x128 prenet out
  float* t_bank = ws + 3145728;             // 8192x2048 conv bank
  float* t_pj   = ws + 19922944;            // 8192x128 proj1 out
  float* t_h    = ws + 20971520;            // 8192x128 highway state
  float* t_A    = t_h1;                     // overlay (h1 dead after prenet)
  float* t_T    = t_h1 + 1048576;
  float* t_mem  = t_bank;                   // overlay (bank dead after proj1)
  float* t_keys = t_bank + 2097152;
  float* t_h0   = t_bank + 4194304;
  float* t_outs = t_h0 + 8192;              // 100x32x63

  auto gemm = [&](const float* A, int lda, const int* ridx,
                  const float* B, int ldb, const float* bias,
                  float* C, int ldc, int M, int N, int K, int act,
                  const float* res, int ldr) {
    int units = (M >> 4) * (N >> 6);        // 16x64 strip per wave
    gemm_k<<<dim3((units + 3) / 4), dim3(128), 0, stream>>>(
        A, lda, ridx, B, ldb, bias, C, ldc, M, N, K, act, res, ldr);
  };

  // prenet (embedding lookup fused into first GEMM)
  gemm(embed, 256, tok, epW0, 256, epb0, t_h1, 256, 8192, 256, 256, ACT_RELU, nullptr, 0);
  gemm(t_h1, 256, nullptr, epW1, 128, epb1, t_x, 128, 8192, 128, 256, ACT_RELU, nullptr, 0);

  // conv bank k=1..16 -> (8192,2048); 512*(128/64)=1024 units -> 256 blocks
  for (int k = 1; k <= 16; ++k)
    conv_gemm_k<<<dim3(256), dim3(128), 0, stream>>>(
        t_x, 7, bankW[k - 1], k, nullptr,
        t_bank, 2048, (k - 1) * 128, 128, ACT_RELU, nullptr, 0, 0);

  // proj1 (maxpool over (t,t+1) fused into the load path), proj2 + residual
  conv_gemm_k<<<dim3(256), dim3(128), 0, stream>>>(
      t_bank, 11, pj1W, 3, pj1b, t_pj, 128, 0, 128, ACT_RELU, nullptr, 0, 1);
  conv_gemm_k<<<dim3(256), dim3(128), 0, stream>>>(
      t_pj, 7, pj2W, 3, pj2b, t_h, 128, 0, 128, ACT_NONE, t_x, 128, 0);

  // 4 highway layers
  for (int i = 0; i < 4; ++i) {
    gemm(t_h, 128, nullptr, hwp[4 * i + 0], 128, hwp[4 * i + 1],
         t_A, 128, 8192, 128, 128, ACT_RELU, nullptr, 0);
    gemm(t_h, 128, nullptr, hwp[4 * i + 2], 128, hwp[4 * i + 3],
         t_T, 128, 8192, 128, 128, ACT_SIG, nullptr, 0);
    hw_combine_k<<<dim3(4096), dim3(256), 0, stream>>>(t_h, t_A, t_T, 1048576);
  }

  // bidirectional GRU scan (dir = blockIdx.x), memory = [hf|hb]
  GruW wf{gfWg, gfbg, gfWc, gfbc}, wb{gbWg, gbbg, gbWc, gbbc};
  enc_gru_k<<<dim3(2), dim3(1024), 65536, stream>>>(t_h, wf, wb, t_mem);

  // attention precompute
  gemm(t_mem, 256, nullptr, attWm, 256, nullptr, t_keys, 256, 8192, 256, 256, ACT_NONE, nullptr, 0);
  gemm(t_mem + 128, 65536, nullptr, aiW, 256, aib, t_h0, 256, 32, 256, 128, ACT_TANH, nullptr, 0);

  // persistent decoder (100 steps), ~304KB dynamic LDS
  DecP dp;
  dp.keys = t_keys; dp.mem = t_mem; dp.Wq = attWq; dp.v = attv;
  dp.arWg = arWg; dp.arbg = arbg; dp.arWc = arWc; dp.arbc = arbc;
  dp.d0Wg = d0Wg; dp.d0bg = d0bg; dp.d0Wc = d0Wc; dp.d0bc = d0bc;
  dp.d1Wg = d1Wg; dp.d1bg = d1bg; dp.d1Wc = d1Wc; dp.d1bc = d1bc;
  dp.Wp0 = dpW0; dp.bp0 = dpb0; dp.Wp1 = dpW1; dp.bp1 = dpb1;
  dp.Wo = WoW; dp.bo = Wob; dp.Wv = WvW; dp.bv = Wvb;
  dp.h0 = t_h0; dp.outs = t_outs; dp.alphas = (float*)d_out + 1008000;
  decoder_k<<<dim3(1), dim3(1024), 311296, stream>>>(dp);

  // final = repeat(outs, 5)[:500]
  final_repeat_k<<<dim3((1008000 + 255) / 256), dim3(256), 0, stream>>>(
      (float*)d_out, t_outs);
}